// LSTM_3204045603927
// MI455X (gfx1250) — compile-verified
//
#include <hip/hip_runtime.h>
#include <math.h>

// ---------------------------------------------------------------------------
// Two-layer LSTM, batch=1, T=32768.  f32 WMMA (v_wmma_f32_16x16x4_f32)
// for both the parallel gate-precompute GEMMs and the recurrent matvecs.
// ---------------------------------------------------------------------------

typedef __attribute__((ext_vector_type(2))) float v2f;
typedef __attribute__((ext_vector_type(8))) float v8f;

#define T_SEQ 32768
#define IN1   128
#define HID   512
#define G1    2048   // 4*HID gates, layer 1
#define G2    512    // 4*IN1 gates, layer 2

__device__ __forceinline__ float sigmoidf_(float x) {
    return 1.0f / (1.0f + __expf(-x));
}

// ---------------------------------------------------------------------------
// C[M,N] = A[M,K] @ W[N,K]^T + (b0[N] + b1[N])
// One wave computes a 64x64 C macro-tile: 4 A-frags x 4 B-frags -> 16 WMMAs
// per K-chunk (4x operand reuse in registers vs the 16x16 version, which
// was L2-read bound: ~2GB+2GB of A/W re-reads; this brings the kernel back
// to the 256MB HBM-write roofline, ~11us at 23.3 TB/s).
// Frag layouts (f32 16x16x4):
//   A: lane<16 holds row (lane&15), K pair {kb,kb+1}; lane>=16 {kb+2,kb+3}.
//   D: VGPR r, lane<16 -> (M=r, N=lane); lane>=16 -> (M=8+r, N=lane-16).
// ---------------------------------------------------------------------------
__global__ __launch_bounds__(256) void wmma_gemm_bias(
    const float* __restrict__ A,      // [M,K]
    const float* __restrict__ W,      // [N,K]
    const float* __restrict__ b0,
    const float* __restrict__ b1,
    float* __restrict__ C,            // [M,N]
    int M, int N, int K)
{
    const int lane = threadIdx.x & 31;
    const int wave = threadIdx.x >> 5;
    const int tile = blockIdx.x * 8 + wave;
    const int tn   = N >> 6;                    // 64-wide macro tiles
    const int mt   = tile / tn;
    const int nt   = tile % tn;

    const int lm   = lane & 15;
    const int kofs = (lane >> 4) << 1;          // 0 or 2

    const float* ap[4];
    const float* wp[4];
#pragma unroll
    for (int i = 0; i < 4; ++i) {
        ap[i] = A + (size_t)(mt * 64 + i * 16 + lm) * K + kofs;
        wp[i] = W + (size_t)(nt * 64 + i * 16 + lm) * K + kofs;
    }

    v8f acc[4][4] = {};
    for (int kb = 0; kb < K; kb += 4) {
        v2f a[4], b[4];
#pragma unroll
        for (int i = 0; i < 4; ++i) {
            a[i] = *(const v2f*)(ap[i] + kb);
            b[i] = *(const v2f*)(wp[i] + kb);
        }
#pragma unroll
        for (int i = 0; i < 4; ++i)
#pragma unroll
            for (int j = 0; j < 4; ++j)
                acc[i][j] = __builtin_amdgcn_wmma_f32_16x16x4_f32(
                    false, a[i], false, b[j], (short)0, acc[i][j], false, false);
    }

#pragma unroll
    for (int j = 0; j < 4; ++j) {
        const int   col  = nt * 64 + j * 16 + lm;
        const float bias = b0[col] + b1[col];
#pragma unroll
        for (int i = 0; i < 4; ++i) {
            const int mbase = mt * 64 + i * 16 + ((lane >> 4) << 3);
            float* cp = C + (size_t)mbase * N + col;
#pragma unroll
            for (int r = 0; r < 8; ++r)
                cp[(size_t)r * N] = acc[i][j][r] + bias;
        }
    }
}

// ---------------------------------------------------------------------------
// Layer-1 recurrence: 16 persistent workgroups, each owns 32 hidden units
// (128 gate rows).  W_hh slice lives in LDS (128 x 516 f32, padded stride
// so the 16-lane fragment column loads hit 16 distinct banks).  Per step:
// stage h (512 f32) to LDS, 8 waves each do a 16-row WMMA matvec over K=512
// using 4 interleaved accumulator chains (breaks the serial D->C WMMA
// dependency that otherwise sets the per-step latency), elementwise cell
// update, write h slice, grid-sync via a monotonic L2 atomic counter.
// Double-buffered h avoids cross-block read/write races.
// ---------------------------------------------------------------------------
__global__ __launch_bounds__(256) void lstm_rec_l1(
    const float* __restrict__ W_hh,   // [2048, 512]
    const float* __restrict__ xg,     // [T, 2048]  (includes both biases)
    float* __restrict__ h_seq,        // [T, 512]
    float* __restrict__ h_db,         // [2][512], zero-initialized
    unsigned* __restrict__ syncp)     // zero-initialized
{
    extern __shared__ float smem[];
    const int WS = 516;                       // padded row stride (banks)
    float* Ws   = smem;                       // 128 x 516
    float* hbuf = Ws + 128 * WS;              // 512
    float* gbuf = hbuf + 512;                 // 128 (i,f,g,o x32)
    float* cbuf = gbuf + 128;                 // 32

    const int tid   = threadIdx.x;
    const int lane  = tid & 31;
    const int wave  = tid >> 5;
    const int ubase = blockIdx.x * 32;        // hidden units owned

    // Stage W slice: LDS row r -> global gate row (r>>5)*512 + ubase + (r&31)
    for (int idx = tid; idx < 128 * 512; idx += 256) {
        int r = idx >> 9, c = idx & 511;
        int grow = ((r >> 5) * HID) + ubase + (r & 31);
        Ws[r * WS + c] = W_hh[(size_t)grow * HID + c];
    }
    if (tid < 32) cbuf[tid] = 0.0f;
    __syncthreads();

    const int kofs = (lane >> 4) << 1;
    const float* awp = Ws + (wave * 16 + (lane & 15)) * WS + kofs;
    const float* hp  = hbuf + kofs;
    const unsigned nblk = gridDim.x;

    for (int t = 0; t < T_SEQ; ++t) {
        const float* hr = h_db + (t & 1) * HID;        // read buffer
        float*       hw = h_db + ((t & 1) ^ 1) * HID;  // write buffer

        for (int i = tid; i < HID; i += 256) hbuf[i] = hr[i];
        __syncthreads();

        v8f a0 = {}, a1 = {}, a2 = {}, a3 = {};        // 4 acc chains
        for (int kb = 0; kb < HID; kb += 16) {
            v2f aa0 = *(const v2f*)(awp + kb);
            v2f bb0 = *(const v2f*)(hp + kb);
            v2f aa1 = *(const v2f*)(awp + kb + 4);
            v2f bb1 = *(const v2f*)(hp + kb + 4);
            v2f aa2 = *(const v2f*)(awp + kb + 8);
            v2f bb2 = *(const v2f*)(hp + kb + 8);
            v2f aa3 = *(const v2f*)(awp + kb + 12);
            v2f bb3 = *(const v2f*)(hp + kb + 12);
            a0 = __builtin_amdgcn_wmma_f32_16x16x4_f32(false, aa0, false, bb0,
                                                       (short)0, a0, false, false);
            a1 = __builtin_amdgcn_wmma_f32_16x16x4_f32(false, aa1, false, bb1,
                                                       (short)0, a1, false, false);
            a2 = __builtin_amdgcn_wmma_f32_16x16x4_f32(false, aa2, false, bb2,
                                                       (short)0, a2, false, false);
            a3 = __builtin_amdgcn_wmma_f32_16x16x4_f32(false, aa3, false, bb3,
                                                       (short)0, a3, false, false);
        }
        v8f acc = (a0 + a1) + (a2 + a3);

        // Column 0 of D lives in lanes 0 (M=0..7) and 16 (M=8..15).
        if ((lane & 15) == 0) {
            int rb = wave * 16 + ((lane >> 4) << 3);
#pragma unroll
            for (int r = 0; r < 8; ++r) {
                int lr   = rb + r;                                  // LDS order
                int grow = ((lr >> 5) * HID) + ubase + (lr & 31);   // global gate
                gbuf[lr] = acc[r] + xg[(size_t)t * G1 + grow];
            }
        }
        __syncthreads();

        if (tid < 32) {
            float iv = sigmoidf_(gbuf[tid]);
            float fv = sigmoidf_(gbuf[32 + tid]);
            float gv = tanhf(gbuf[64 + tid]);
            float ov = sigmoidf_(gbuf[96 + tid]);
            float c  = fv * cbuf[tid] + iv * gv;
            cbuf[tid] = c;
            float h = ov * tanhf(c);
            hw[ubase + tid] = h;
            h_seq[(size_t)t * HID + ubase + tid] = h;
        }
        __syncthreads();

        // Grid sync: monotonic counter, no reset (32768*16 < 2^32).
        if (tid == 0) {
            __threadfence();
            __hip_atomic_fetch_add(syncp, 1u, __ATOMIC_ACQ_REL,
                                   __HIP_MEMORY_SCOPE_AGENT);
            unsigned target = (unsigned)(t + 1) * nblk;
            while (__hip_atomic_load(syncp, __ATOMIC_ACQUIRE,
                                     __HIP_MEMORY_SCOPE_AGENT) < target) {
                __builtin_amdgcn_s_sleep(2);
            }
        }
        __syncthreads();
    }
}

// ---------------------------------------------------------------------------
// Layer-2 recurrence: W_hh2 (512x128 f32 = 256KB) fits in ONE workgroup's
// 320KB LDS -> single persistent block, h/c never leave LDS, only
// workgroup barriers per step.  8 waves x 4 tiles cover the 512 gate rows;
// the 4 per-tile accumulators already interleave the WMMA chains.
// ---------------------------------------------------------------------------
__global__ __launch_bounds__(256) void lstm_rec_l2(
    const float* __restrict__ W_hh,   // [512, 128]
    const float* __restrict__ xg,     // [T, 512]
    float* __restrict__ out)          // [T, 128]
{
    extern __shared__ float smem[];
    const int WS = 132;                       // padded row stride
    float* Ws   = smem;                       // 512 x 132
    float* hbuf = Ws + 512 * WS;              // 128
    float* gbuf = hbuf + 128;                 // 512
    float* cbuf = gbuf + 512;                 // 128

    const int tid = threadIdx.x, lane = tid & 31, wave = tid >> 5;

    for (int idx = tid; idx < 512 * 128; idx += 256) {
        int r = idx >> 7, c = idx & 127;
        Ws[r * WS + c] = W_hh[idx];
    }
    if (tid < 128) { hbuf[tid] = 0.0f; cbuf[tid] = 0.0f; }
    __syncthreads();

    const int kofs = (lane >> 4) << 1;
    const float* awp = Ws + (wave * 64 + (lane & 15)) * WS + kofs;
    const float* hp  = hbuf + kofs;

    for (int t = 0; t < T_SEQ; ++t) {
        v8f acc[4] = {};
#pragma unroll 4
        for (int kb = 0; kb < IN1; kb += 4) {
            v2f b = *(const v2f*)(hp + kb);
#pragma unroll
            for (int j = 0; j < 4; ++j) {
                v2f a = *(const v2f*)(awp + j * 16 * WS + kb);
                acc[j] = __builtin_amdgcn_wmma_f32_16x16x4_f32(false, a, false, b,
                                                  (short)0, acc[j], false, false);
            }
        }
        if ((lane & 15) == 0) {
            const int hi = (lane >> 4) << 3;
#pragma unroll
            for (int j = 0; j < 4; ++j) {
                int rb = wave * 64 + j * 16 + hi;
#pragma unroll
                for (int r = 0; r < 8; ++r)
                    gbuf[rb + r] = acc[j][r] + xg[(size_t)t * G2 + rb + r];
            }
        }
        __syncthreads();

        if (tid < 128) {
            float iv = sigmoidf_(gbuf[tid]);
            float fv = sigmoidf_(gbuf[128 + tid]);
            float gv = tanhf(gbuf[256 + tid]);
            float ov = sigmoidf_(gbuf[384 + tid]);
            float c  = fv * cbuf[tid] + iv * gv;
            cbuf[tid] = c;
            float h = ov * tanhf(c);
            hbuf[tid] = h;
            out[(size_t)t * IN1 + tid] = h;
        }
        __syncthreads();
    }
}

// ---------------------------------------------------------------------------
extern "C" void kernel_launch(void* const* d_in, const int* in_sizes, int n_in,
                              void* d_out, int out_size, void* d_ws, size_t ws_size,
                              hipStream_t stream)
{
    (void)in_sizes; (void)n_in; (void)out_size; (void)ws_size;

    const float* X     = (const float*)d_in[0];
    const float* W_ih1 = (const float*)d_in[1];
    const float* W_hh1 = (const float*)d_in[2];
    const float* b_ih1 = (const float*)d_in[3];
    const float* b_hh1 = (const float*)d_in[4];
    const float* W_ih2 = (const float*)d_in[5];
    const float* W_hh2 = (const float*)d_in[6];
    const float* b_ih2 = (const float*)d_in[7];
    const float* b_hh2 = (const float*)d_in[8];
    float* out = (float*)d_out;

    // Workspace layout (assumes ws_size >= ~390 MB):
    //   [0, 256)        sync counter
    //   [256, 4352)     h double buffer (2 x 512 f32), zero = h0
    //   [8192, ...)     xg1 [T,2048] f32 | h1 [T,512] f32 | xg2 [T,512] f32
    char*     ws    = (char*)d_ws;
    unsigned* syncp = (unsigned*)ws;
    float*    h_db  = (float*)(ws + 256);
    float*    xg1   = (float*)(ws + 8192);
    float*    h1    = xg1 + (size_t)T_SEQ * G1;
    float*    xg2   = h1  + (size_t)T_SEQ * HID;

    hipMemsetAsync(ws, 0, 8192, stream);   // zero sync counter + h0 buffers

    // Layer 1 gate precompute: [32768,128] x [2048,128]^T, 64x64 wave tiles
    wmma_gemm_bias<<<(T_SEQ / 64) * (G1 / 64) / 8, 256, 0, stream>>>(
        X, W_ih1, b_ih1, b_hh1, xg1, T_SEQ, G1, IN1);

    // Layer 1 recurrence: 16 blocks, W_hh1 slice in LDS (516-float stride)
    size_t lds1 = (size_t)(128 * 516 + 512 + 128 + 32) * sizeof(float);
    lstm_rec_l1<<<16, 256, lds1, stream>>>(W_hh1, xg1, h1, h_db, syncp);

    // Layer 2 gate precompute: [32768,512] x [512,512]^T, 64x64 wave tiles
    wmma_gemm_bias<<<(T_SEQ / 64) * (G2 / 64) / 8, 256, 0, stream>>>(
        h1, W_ih2, b_ih2, b_hh2, xg2, T_SEQ, G2, HID);

    // Layer 2 recurrence: single block, everything LDS-resident
    size_t lds2 = (size_t)(512 * 132 + 128 + 512 + 128) * sizeof(float);
    lstm_rec_l2<<<1, 256, lds2, stream>>>(W_hh2, xg2, out);
}